// EmbeddingLoss_71133248356357
// MI455X (gfx1250) — compile-verified
//
#include <hip/hip_runtime.h>
#include <hip/hip_bf16.h>
#include <math.h>

// ---------------------------------------------------------------------------
// EmbeddingLoss (discriminative instance loss + BCE) for MI455X / gfx1250.
//
// Pass 1 (stats_kernel): streaming segmented reduction over ~270MB,
//   implemented with V_WMMA_F32_16X16X4_F32 (full f32 precision):
//     sums[32,16] = (w*OneHot)[32,N] x Emb[N,16]   (2 WMMA tiles / 4 voxels)
//     sq-sums     = (w*OneHot) x Emb.^2            (2 WMMA tiles / 4 voxels)
//     cnt         = per-lane scalar accumulation of the A-tile entries
//   plus a fused BCE-with-logits partial sum. Roofline: ~270MB @ 23.3TB/s
//   => ~12us floor; the loop carries no LDS traffic (vmem + VALU + WMMA only).
// Pass 2 (finalize_kernel): 1 block, 4 warps (one per batch): centers,
//   internal/external/reg terms, 32x32 hinge via shuffles, BCE mean, total.
// ---------------------------------------------------------------------------

#define SDIM   96
#define NVOX   (SDIM * SDIM * SDIM)   // 884736, divisible by 32
#define ECH    16
#define KLAB   32
#define BATCH  4
#define DELTA_ 1.5f

// workspace layout (floats): per batch 576 = [32 clusters x 16 sums][32 cnt][32 sq]
// stats[b*576 + m*16 + e] : sums ; stats[b*576 + 512 + m] : cnt ; stats[b*576 + 544 + m] : sq
// stats[2304] : global BCE sum
#define WS_FLOATS 2305

typedef float v2f __attribute__((ext_vector_type(2)));
typedef float v8f __attribute__((ext_vector_type(8)));

__global__ __launch_bounds__(256) void stats_kernel(
    const float* __restrict__ emb,     // [B, E, N]
    const float* __restrict__ logits,  // [B, 1, N]
    const int*   __restrict__ lbl,     // [B, 1, N]
    const float* __restrict__ tmsk,    // [B, 1, N]
    float* __restrict__ stats)
{
    const int b = blockIdx.y;
    const size_t embBase = (size_t)b * ECH * NVOX;
    const size_t vBase   = (size_t)b * NVOX;

    const int lane  = threadIdx.x & 31;
    const int half  = lane >> 4;       // 0: voxel k-slots {0,1} ; 1: k-slots {2,3}
    const int c     = lane & 15;       // channel (B tile) / cluster-in-half (A tile)
    const int wid    = (blockIdx.x * blockDim.x + threadIdx.x) >> 5;
    const int nWaves = (gridDim.x * blockDim.x) >> 5;

    __shared__ float lsum[KLAB * ECH];  // 512
    __shared__ float lcnt[KLAB];
    __shared__ float lsq[KLAB];
    __shared__ float lbce;
    for (int i = threadIdx.x; i < KLAB * ECH; i += blockDim.x) lsum[i] = 0.0f;
    if (threadIdx.x < KLAB) { lcnt[threadIdx.x] = 0.0f; lsq[threadIdx.x] = 0.0f; }
    if (threadIdx.x == 0) lbce = 0.0f;

    // WMMA accumulators: [clusters 0..15 | 16..31] x {sums, sq}
    v8f acc0 = {}; v8f acc1 = {};
    v8f asq0 = {}; v8f asq1 = {};
    // per-lane scalar count accumulators (cluster c and cluster c+16,
    // restricted to this lane's half of the voxels; combined at the end)
    float cnt0 = 0.0f, cnt1 = 0.0f;

    const float* ep   = emb + embBase + (size_t)c * NVOX; // this lane's channel row
    const int*   lp   = lbl + vBase;
    const float* wp   = tmsk + vBase;

    const int nChunks = NVOX / 32;
    for (int chunk = wid; chunk < nChunks; chunk += nWaves) {
        const int vb = chunk * 32;
        #pragma unroll
        for (int s = 0; s < 8; ++s) {
            const int kv = vb + s * 4 + half * 2;  // this lane's first voxel (k slot)
            // B tile: embedding values, this lane's channel, two voxels -> b64
            float b0 = ep[kv];
            float b1 = ep[kv + 1];
            // A tile inputs: label + mask of the same two voxels (broadcast
            // b64 loads; whole 128B line consumed by the wave across subs)
            int   lA = lp[kv];
            int   lB = lp[kv + 1];
            float tA = wp[kv];
            float tB = wp[kv + 1];
            const bool mA = (tA != 0.0f);
            const bool mB = (tB != 0.0f);
            v2f a0 = { (mA && lA == c)      ? 1.0f : 0.0f,
                       (mB && lB == c)      ? 1.0f : 0.0f };
            v2f a1 = { (mA && lA == c + 16) ? 1.0f : 0.0f,
                       (mB && lB == c + 16) ? 1.0f : 0.0f };
            v2f bt = { b0, b1 };
            v2f bs = { b0 * b0, b1 * b1 };
            acc0 = __builtin_amdgcn_wmma_f32_16x16x4_f32(false, a0, false, bt, (short)0, acc0, false, false);
            acc1 = __builtin_amdgcn_wmma_f32_16x16x4_f32(false, a1, false, bt, (short)0, acc1, false, false);
            asq0 = __builtin_amdgcn_wmma_f32_16x16x4_f32(false, a0, false, bs, (short)0, asq0, false, false);
            asq1 = __builtin_amdgcn_wmma_f32_16x16x4_f32(false, a1, false, bs, (short)0, asq1, false, false);
            cnt0 += a0.x + a0.y;
            cnt1 += a1.x + a1.y;
        }
    }

    // fused BCE-with-logits partial sum over this batch's logits
    float bacc = 0.0f;
    {
        const int tid = blockIdx.x * blockDim.x + threadIdx.x;
        const int tot = gridDim.x * blockDim.x;
        for (int i = tid; i < NVOX; i += tot) {
            float x = logits[vBase + i];
            float t = wp[i];
            bacc += fmaxf(x, 0.0f) - x * t + log1pf(__expf(-fabsf(x)));
        }
        #pragma unroll
        for (int off = 16; off > 0; off >>= 1) bacc += __shfl_xor(bacc, off, 32);
    }

    __syncthreads();   // LDS init visible to all before atomic phase

    // D layout: lane L, vgpr r holds D[M = r + 8*(L>=16)][N = L&15]
    #pragma unroll
    for (int r = 0; r < 8; ++r) {
        const int m = r + 8 * half;
        atomicAdd(&lsum[m * ECH + c],         acc0[r]);
        atomicAdd(&lsum[(m + 16) * ECH + c],  acc1[r]);
        // sq needs row-sum over N (the 16 lanes of this half-wave)
        float v0 = asq0[r], v1 = asq1[r];
        #pragma unroll
        for (int off = 1; off <= 8; off <<= 1) {
            v0 += __shfl_xor(v0, off, 32);
            v1 += __shfl_xor(v1, off, 32);
        }
        if (c == 0) {   // lanes 0 and 16
            atomicAdd(&lsq[m],       v0);
            atomicAdd(&lsq[m + 16],  v1);
        }
    }
    // counts: combine the two halves (each saw disjoint voxel k-slots)
    cnt0 += __shfl_xor(cnt0, 16, 32);
    cnt1 += __shfl_xor(cnt1, 16, 32);
    if (half == 0) {
        atomicAdd(&lcnt[c],      cnt0);
        atomicAdd(&lcnt[c + 16], cnt1);
    }
    if (lane == 0) atomicAdd(&lbce, bacc);

    __syncthreads();

    float* sb = stats + (size_t)b * 576;
    for (int i = threadIdx.x; i < KLAB * ECH; i += blockDim.x) atomicAdd(&sb[i], lsum[i]);
    if (threadIdx.x < KLAB) {
        atomicAdd(&sb[512 + threadIdx.x], lcnt[threadIdx.x]);
        atomicAdd(&sb[544 + threadIdx.x], lsq[threadIdx.x]);
    }
    if (threadIdx.x == 0) atomicAdd(&stats[2304], lbce);
}

__device__ __forceinline__ float warpSum32(float v) {
    #pragma unroll
    for (int off = 16; off > 0; off >>= 1) v += __shfl_xor(v, off, 32);
    return v;
}

__global__ __launch_bounds__(128) void finalize_kernel(
    const float* __restrict__ stats, float* __restrict__ out)
{
    const int b = threadIdx.x >> 5;     // warp = batch
    const int m = threadIdx.x & 31;     // lane = cluster
    const float* sb = stats + b * 576;

    float cnt  = sb[512 + m];
    float sq   = sb[544 + m];
    float safe = fmaxf(cnt, 1.0f);

    float center[ECH];
    float cent_sq = 0.0f;
    #pragma unroll
    for (int e = 0; e < ECH; ++e) {
        float ce = sb[m * ECH + e] / safe;
        center[e] = ce;
        cent_sq += ce * ce;
    }

    const float fgf = ((cnt > 0.0f) && (m > 0)) ? 1.0f : 0.0f;
    const float C = warpSum32(fgf);
    const float Csafe = fmaxf(C, 1.0f);

    float internal = warpSum32(fgf * (sq / safe - cent_sq));
    internal = (C >= 1.0f) ? internal / Csafe : 0.0f;

    float reg = warpSum32(fgf * sqrtf(cent_sq));
    reg = (C >= 1.0f) ? reg / Csafe : 0.0f;

    // pairwise hinge on center distances
    float hs = 0.0f;
    for (int j = 0; j < KLAB; ++j) {
        float d2 = 0.0f;
        #pragma unroll
        for (int e = 0; e < ECH; ++e) {
            float cj = __shfl(center[e], j, 32);
            float df = center[e] - cj;
            d2 += df * df;
        }
        float fgj = __shfl(fgf, j, 32);
        if ((fgf > 0.5f) && (fgj > 0.5f) && (j != m)) {
            float d = sqrtf(d2);
            float h = fmaxf(2.0f * DELTA_ - d, 0.0f);
            hs += h * h;
        }
    }
    float external = warpSum32(hs);
    external = (C >= 2.0f) ? external / fmaxf(C * (C - 1.0f), 1.0f) : 0.0f;

    __shared__ float ri[BATCH], re[BATCH], rr[BATCH];
    if (m == 0) { ri[b] = internal; re[b] = external; rr[b] = reg; }
    __syncthreads();
    if (threadIdx.x == 0) {
        float mi = 0.0f, me = 0.0f, mr = 0.0f;
        for (int i = 0; i < BATCH; ++i) { mi += ri[i]; me += re[i]; mr += rr[i]; }
        mi *= (1.0f / BATCH); me *= (1.0f / BATCH); mr *= (1.0f / BATCH);
        float bce = stats[2304] / ((float)BATCH * (float)NVOX);
        // total = ALPHA*mean(internal) + BETA*mean(external) + GAMMA + mean(reg) + bce
        out[0] = 1.0f * mi + 1.0f * me + 0.001f + mr + bce;
    }
}

extern "C" void kernel_launch(void* const* d_in, const int* in_sizes, int n_in,
                              void* d_out, int out_size, void* d_ws, size_t ws_size,
                              hipStream_t stream) {
    const float* emb    = (const float*)d_in[0];   // [4,16,96,96,96] f32
    const float* logits = (const float*)d_in[1];   // [4,1,96,96,96]  f32
    const int*   lbl    = (const int*)d_in[2];     // [4,1,96,96,96]  i32
    const float* tmsk   = (const float*)d_in[3];   // [4,1,96,96,96]  f32
    float* out   = (float*)d_out;
    float* stats = (float*)d_ws;
    (void)in_sizes; (void)n_in; (void)out_size; (void)ws_size;

    hipMemsetAsync(stats, 0, WS_FLOATS * sizeof(float), stream);

    dim3 grid(128, BATCH);   // 512 blocks x 256 thr = 4096 waves; mem-bound streaming
    stats_kernel<<<grid, 256, 0, stream>>>(emb, logits, lbl, tmsk, stats);
    finalize_kernel<<<1, 128, 0, stream>>>(stats, out);
}